// GroupNorm2dInteger_57930518889080
// MI455X (gfx1250) — compile-verified
//
#include <hip/hip_runtime.h>
#include <math.h>

// ---------------------------------------------------------------------------
// GroupNorm2dInteger for x: (64, 128, 64, 64) fp32.
//   Pass A: per-(batch, chunk) partial sum / sumsq   (reads 128 MiB from HBM)
//   Pass B: 64-thread stats kernel: mu_q, fixed-point isqrt via LUT
//   Pass C: elementwise normalize + Q8.4 floor-cast  (input re-read hits L2,
//           256 MiB of outputs streamed with non-temporal stores)
// Memory-bound: ~384 MiB of HBM traffic => ~17 us floor at 23.3 TB/s.
// Cross-lane reduction uses V_WMMA_F32_16X16X4_F32 (exact fp32 pair-adds).
// ---------------------------------------------------------------------------

typedef float  v2f  __attribute__((ext_vector_type(2)));
typedef float  v4f  __attribute__((ext_vector_type(4)));
typedef float  v8f  __attribute__((ext_vector_type(8)));
typedef int    v4i  __attribute__((ext_vector_type(4)));

#define NBATCH      64
#define PER_BATCH   524288      // 128*64*64
#define NBLK        128         // chunks per batch
#define BLOCK       256         // threads per block (8 waves)
#define CHUNK       4096        // elements per block = PER_BATCH / NBLK
#define SQRT2_FIX   46340       // floor(sqrt(2) * 2^15)

// ---- Pass A: partial (sum, sumsq) per (batch, chunk) ----------------------
__global__ __launch_bounds__(BLOCK) void gni_reduce(
    const float* __restrict__ x, float2* __restrict__ partials)
{
    const int batch = blockIdx.y;
    const int blk   = blockIdx.x;
    const v4f* p = (const v4f*)(x + (size_t)batch * PER_BATCH
                                  + (size_t)blk   * CHUNK);

    float s = 0.0f, q = 0.0f;
#pragma unroll
    for (int i = 0; i < 4; ++i) {
        v4f v = p[threadIdx.x + i * BLOCK];
        s += (v.x + v.y) + (v.z + v.w);
        q = fmaf(v.x, v.x, q); q = fmaf(v.y, v.y, q);
        q = fmaf(v.z, v.z, q); q = fmaf(v.w, v.w, q);
    }

    // ---- wave32 reduction via V_WMMA_F32_16X16X4_F32 ----------------------
    // A (16x4): row0 = [1,1,0,0], row1 = [0,0,1,1], rest 0.
    //   layout: lane L<16 holds A[L,0](v0),A[L,1](v1); lane L>=16 holds
    //           A[L-16,2](v0),A[L-16,3](v1)  -> ones in lanes 0 and 17 only.
    // B (4x16): v0 = {B[0,n] lanes0-15, B[1,n] lanes16-31} = lane sums,
    //           v1 = {B[2,n], B[3,n]}                      = lane sumsqs.
    // D[0,n] = sum_n + sum_{n+16} ; D[1,n] = sq_n + sq_{n+16}
    const unsigned lane = threadIdx.x & 31u;
    v2f a;
    a.x = (lane == 0u || lane == 17u) ? 1.0f : 0.0f;
    a.y = a.x;
    v2f b;
    b.x = s;
    b.y = q;
    v8f c = {};
    v8f d = __builtin_amdgcn_wmma_f32_16x16x4_f32(
        /*neg_a=*/false, a, /*neg_b=*/false, b,
        /*c_mod=*/(short)0, c, /*reuse_a=*/false, /*reuse_b=*/false);

    float s2 = d[0];   // row 0 in lanes 0-15: pairwise sums
    float q2 = d[1];   // row 1 in lanes 0-15: pairwise square-sums
#pragma unroll
    for (int off = 8; off >= 1; off >>= 1) {
        s2 += __shfl_xor(s2, off, 32);
        q2 += __shfl_xor(q2, off, 32);
    }

    __shared__ float2 red[BLOCK / 32];
    if (lane == 0) red[threadIdx.x >> 5] = make_float2(s2, q2);
    __syncthreads();

    if (threadIdx.x == 0) {
        float ss = 0.0f, qq = 0.0f;
#pragma unroll
        for (int w = 0; w < BLOCK / 32; ++w) { ss += red[w].x; qq += red[w].y; }
        partials[batch * NBLK + blk] = make_float2(ss, qq);
    }
}

// ---- Pass B: per-batch stats (one thread per batch) -----------------------
__global__ __launch_bounds__(64) void gni_stats(
    const float2* __restrict__ partials, const int* __restrict__ lut,
    float2* __restrict__ stats)
{
    const int bt = threadIdx.x;           // 64 threads, batch = thread
    double s = 0.0, q = 0.0;
    for (int i = 0; i < NBLK; ++i) {
        float2 p = partials[bt * NBLK + i];
        s += (double)p.x;
        q += (double)p.y;
    }
    const double N = (double)PER_BATCH;
    const double mu = s / N;

    // integer_floor_quantizer(mu, W=8, F=4)
    double mf = floor(mu * 16.0);
    mf = fmin(fmax(mf, -128.0), 127.0);
    const double mu_q = mf / 16.0;

    // var = E[x^2] - 2*mu_q*E[x] + mu_q^2  (exact vs reference up to fp assoc.)
    double var = q / N - 2.0 * mu_q * mu + mu_q * mu_q;
    if (var < 0.0) var = 0.0;

    // integer_floor_quantizer(var, W=16, F=8) -> var_int (>= 0, <= 32767)
    double vf = floor(var * 256.0);
    if (vf > 32767.0) vf = 32767.0;
    int var_int = (int)vf;

    // fixed-point inverse sqrt, W=16, F=8, P=5
    int res;
    if (var_int == 0) {
        res = 65535;                                  // saturate on zero var
    } else {
        int msb   = 31 - __clz(var_int);              // highest set bit
        int x_red = var_int << (15 - msb);            // align MSB to bit 15
        int idx   = (x_red >> 10) & 31;               // top 5 fraction bits
        int lv    = lut[idx];
        int e     = 24 - msb;                         // 3*F - msb
        if (e & 1) lv = (lv * SQRT2_FIX) >> 15;
        int e2 = e >> 1;                              // e >= 10 here
        int sh = 15 - e2;
        res = (sh >= 0) ? (lv >> sh) : (lv << (-sh));
        if (res < 0) res = 0;
        if (res > 65535) res = 65535;
    }
    // integer_floor_quantizer(res/256, W=16, F=8) == min(res, 32767)/256
    if (res > 32767) res = 32767;
    stats[bt] = make_float2((float)mu_q, (float)res * (1.0f / 256.0f));
}

// ---- Pass C: normalize + Q8.4 floor-cast, NT stores -----------------------
__global__ __launch_bounds__(BLOCK) void gni_norm(
    const float* __restrict__ x, const float2* __restrict__ stats,
    float* __restrict__ outf, int* __restrict__ outi)
{
    const int batch = blockIdx.y;
    const float2 st = stats[batch];
    const float mu  = st.x;
    const float inv = st.y;

    const size_t base = (size_t)batch * PER_BATCH + (size_t)blockIdx.x * CHUNK;
    const v4f* xp = (const v4f*)(x + base);
    v4f*       fp = (v4f*)(outf + base);
    v4i*       ip = (v4i*)(outi + base);

#pragma unroll
    for (int i = 0; i < 4; ++i) {
        const int off = threadIdx.x + i * BLOCK;
        v4f v = xp[off];
        v4f fo;
        v4i io;
#pragma unroll
        for (int j = 0; j < 4; ++j) {
            float t  = (v[j] - mu) * inv;
            float xi = floorf(t * 16.0f);
            xi = fminf(fmaxf(xi, -128.0f), 127.0f);
            fo[j] = xi * 0.0625f;
            io[j] = ((int)xi) & 255;      // two's-complement bits mod 2^8
        }
        __builtin_nontemporal_store(fo, fp + off);
        __builtin_nontemporal_store(io, ip + off);
    }
}

// ---------------------------------------------------------------------------
extern "C" void kernel_launch(void* const* d_in, const int* in_sizes, int n_in,
                              void* d_out, int out_size, void* d_ws, size_t ws_size,
                              hipStream_t stream)
{
    const float* x   = (const float*)d_in[0];
    const int*   lut = (const int*)d_in[1];

    const size_t nelem = (size_t)NBATCH * PER_BATCH;   // 33,554,432
    float* outf = (float*)d_out;                        // output 0: norm_float
    int*   outi = (int*)d_out + nelem;                  // output 1: norm_uint

    float2* partials = (float2*)d_ws;                           // 8192 * 8 B
    float2* stats    = (float2*)((char*)d_ws +
                                 (size_t)NBATCH * NBLK * sizeof(float2));

    dim3 grid(NBLK, NBATCH), block(BLOCK);
    gni_reduce<<<grid, block, 0, stream>>>(x, partials);
    gni_stats<<<1, 64, 0, stream>>>(partials, lut, stats);
    gni_norm<<<grid, block, 0, stream>>>(x, stats, outf, outi);
}